// KalmanNetV1_67637144977675
// MI455X (gfx1250) — compile-verified
//
#include <hip/hip_runtime.h>

typedef __attribute__((ext_vector_type(2))) float v2f;
typedef __attribute__((ext_vector_type(4))) float v4f;
typedef __attribute__((ext_vector_type(8))) float v8f;

#define KF_T      8388608
#define KF_T0     256                                // sequential prefix (Riccati converges in <64 steps)
#define KF_TILE   256                                // samples per WMMA tile (16 segs x 16)
#define KF_TPC    16                                 // tiles per chunk (wave) -> 4096 samples
#define KF_TILES  ((KF_T - KF_T0) / KF_TILE)         // 32767
#define KF_CHUNKS ((KF_TILES + KF_TPC - 1) / KF_TPC) // 2048 (last chunk: 15 tiles)

// workspace layout (floats)
#define WS_XINIT 0
#define WS_A     1
#define WS_B     2
#define WS_C     16                     // per-chunk aggregate C (2048)
#define WS_AAGG  (16 + KF_CHUNKS)       // per-chunk aggregate A (2048)
#define WS_TIN   (16 + 2 * KF_CHUNKS)   // per-chunk incoming carry (2048)

// a^e via masked uniform loop (register-only, e in [0,16])
__device__ __forceinline__ float kf_pow(float a, int e) {
    float p = 1.0f;
#pragma unroll
    for (int i = 0; i < 16; ++i) p = (i < e) ? p * a : p;
    return p;
}

// ---------------- Kernel 1: exact sequential prefix, derive steady coeffs ----------------
__global__ void kf_seq_head(const float* __restrict__ x,
                            const float* x0p, const float* p0p,
                            const float* Ap, const float* Hp,
                            const float* Qp, const float* Rp,
                            float* __restrict__ out, float* __restrict__ ws) {
    if (threadIdx.x != 0 || blockIdx.x != 0) return;
    float xs = x0p[0], ps = p0p[0];
    const float A = Ap[0], H = Hp[0], Q = Qp[0], R = Rp[0];
    for (int t = 0; t < KF_T0; ++t) {
        float y  = x[t];
        float xp = A * xs;
        float Pp = A * ps * A + Q;
        float K  = Pp * H / (H * Pp * H + R);
        xs = xp + K * (y - H * xp);
        ps = Pp - K * H * Pp;
        out[t] = xs;
    }
    // steady-state gain for t >= KF_T0 (P has reached its f32 fixed point)
    float Pp = A * ps * A + Q;
    float K  = Pp * H / (H * Pp * H + R);
    ws[WS_XINIT] = xs;
    ws[WS_A] = (1.0f - K * H) * A;  // x_t = a*x_{t-1} + b*y_t
    ws[WS_B] = K;
}

// ---------------- per-wave lane constants ----------------
struct KfConsts {
    v2f  aMat[4];  // A-operand per K=4 chunk (L lower-tri Toeplitz, b*a^{m-j})
    float pw[8];   // a^{i+1} for this lane's C/D rows
    float a16, a16s, a256;
};

__device__ __forceinline__ void kf_setup(float a, float b, int lane, KfConsts& c) {
    const int m  = lane & 15;
    const int cb = (lane < 16) ? 0 : 2;
#pragma unroll
    for (int k = 0; k < 4; ++k) {
        int j0 = 4 * k + cb;
        c.aMat[k][0] = (j0     <= m) ? b * kf_pow(a, m - j0)     : 0.0f;
        c.aMat[k][1] = (j0 + 1 <= m) ? b * kf_pow(a, m - j0 - 1) : 0.0f;
    }
    const int be = (lane < 16) ? 1 : 9;
#pragma unroll
    for (int v = 0; v < 8; ++v) c.pw[v] = kf_pow(a, be + v);
    c.a16  = kf_pow(a, 16);
    c.a256 = c.a16;
#pragma unroll
    for (int i = 0; i < 4; ++i) c.a256 *= c.a256;  // a16^16 = a^256
    c.a16s = kf_pow(c.a16, m);
}

// ---------------- tile load (B operands for 4 chained K=4 WMMAs) ----------------
struct TileB { v2f b0, b1, b2, b3; };

__device__ __forceinline__ TileB kf_load(const float* __restrict__ yTile, int lane) {
    const float* base = yTile + (lane & 15) * 16 + ((lane < 16) ? 0 : 2);
    TileB t;
    t.b0 = *(const v2f*)(base + 0);
    t.b1 = *(const v2f*)(base + 4);
    t.b2 = *(const v2f*)(base + 8);
    t.b3 = *(const v2f*)(base + 12);
    return t;
}

// ---------------- WMMA tile compute: Out = L(16x16) x Y(16x16), then segment fixup --------
// C/D layout: lane s holds rows 0..7 of column s; lane s+16 holds rows 8..15.
template <bool STORE>
__device__ __forceinline__ float kf_compute(const TileB& tb, float* __restrict__ outTile,
                                            float x_in, const KfConsts& c, int lane) {
    const int s = lane & 15;
    v8f d = {};
    d = __builtin_amdgcn_wmma_f32_16x16x4_f32(false, c.aMat[0], false, tb.b0, (short)0, d, false, false);
    d = __builtin_amdgcn_wmma_f32_16x16x4_f32(false, c.aMat[1], false, tb.b1, (short)0, d, false, false);
    d = __builtin_amdgcn_wmma_f32_16x16x4_f32(false, c.aMat[2], false, tb.b2, (short)0, d, false, false);
    d = __builtin_amdgcn_wmma_f32_16x16x4_f32(false, c.aMat[3], false, tb.b3, (short)0, d, false, false);
    // o_s = Out[15][s] lives in lane 16+s, VGPR 7; replicate to both half-waves.
    float o = __shfl(d[7], 16 + s);
    // weighted inclusive scan: u_s = sum_{r<=s} a16^{s-r} * o_r  (Hillis-Steele)
    float u = o, w = c.a16;
#pragma unroll
    for (int st = 1; st < 16; st <<= 1) {
        float g = __shfl(u, (lane >= st) ? lane - st : lane);
        u += (s >= st) ? w * g : 0.0f;
        w *= w;
    }
    float um1 = __shfl(u, (lane > 0) ? lane - 1 : 0);
    float t   = c.a16s * x_in + ((s == 0) ? 0.0f : um1);  // segment-entering state
    float u15 = __shfl(u, 15);
    if (STORE) {
        v4f lo, hi;
#pragma unroll
        for (int v = 0; v < 4; ++v) lo[v] = d[v] + c.pw[v] * t;
#pragma unroll
        for (int v = 0; v < 4; ++v) hi[v] = d[v + 4] + c.pw[v + 4] * t;
        float* o0 = outTile + s * 16 + ((lane < 16) ? 0 : 8);
        *(v4f*)(o0)     = lo;
        *(v4f*)(o0 + 4) = hi;
    }
    return c.a256 * x_in + u15;  // tile carry-out
}

// ---------------- chunk driver: software-pipelined tile loop (uniform bounds) ----------------
template <bool STORE>
__device__ __forceinline__ float kf_chunk(const float* __restrict__ x, float* __restrict__ out,
                                          int tile0, int ntiles, float carry,
                                          const KfConsts& c, int lane) {
    TileB cur = kf_load(x + KF_T0 + tile0 * KF_TILE, lane);
    for (int g = 0; g < ntiles; ++g) {
        const int gn = (g + 1 < ntiles) ? g + 1 : g;  // clamp: no OOB read on last tile
        TileB nxt = kf_load(x + KF_T0 + (tile0 + gn) * KF_TILE, lane);  // overlaps fixup below
        carry = kf_compute<STORE>(cur, out + KF_T0 + (tile0 + g) * KF_TILE, carry, c, lane);
        cur = nxt;
    }
    return carry;
}

// ---------------- Phase A: per-chunk aggregates (zero initial state) ----------------
__global__ void kf_phaseA(const float* __restrict__ x, float* __restrict__ ws) {
    const int lane  = threadIdx.x & 31;
    const int chunk = __builtin_amdgcn_readfirstlane(blockIdx.x * (blockDim.x >> 5) + (threadIdx.x >> 5));
    const float a = ws[WS_A], b = ws[WS_B];
    KfConsts c;
    kf_setup(a, b, lane, c);
    const int tile0  = chunk * KF_TPC;
    const int ntiles = __builtin_amdgcn_readfirstlane(min(KF_TPC, KF_TILES - tile0));
    float carry = kf_chunk<false>(x, nullptr, tile0, ntiles, 0.0f, c, lane);
    float Aagg = kf_pow(c.a256, ntiles);  // a^(256*ntiles)
    if (lane == 0) {
        ws[WS_C + chunk]    = carry;
        ws[WS_AAGG + chunk] = Aagg;
    }
}

// ---------------- Phase scan: affine scan over 2048 chunk aggregates (1 wave) ----------------
__global__ void kf_scan(float* __restrict__ ws) {
    const int lane = threadIdx.x & 31;
    const int per  = KF_CHUNKS / 32;  // 64 chunks per lane
    const float x_init = ws[WS_XINIT];
    const float* C  = ws + WS_C;
    const float* Ag = ws + WS_AAGG;
    float* tIn = ws + WS_TIN;
    const int base = lane * per;
    float Aacc = 1.0f, Cacc = 0.0f;  // composition of this lane's chunks (applied in order)
    for (int i = 0; i < per; ++i) {
        float Ac = Ag[base + i], Cc = C[base + i];
        Cacc = Ac * Cacc + Cc;
        Aacc = Ac * Aacc;
    }
    // inclusive affine scan across lanes: new = cur o prev
    float Ai = Aacc, Ci = Cacc;
#pragma unroll
    for (int d = 1; d < 32; d <<= 1) {
        float Ap = __shfl(Ai, (lane >= d) ? lane - d : lane);
        float Cp = __shfl(Ci, (lane >= d) ? lane - d : lane);
        float An = Ai * Ap;
        float Cn = Ai * Cp + Ci;
        Ai = (lane >= d) ? An : Ai;
        Ci = (lane >= d) ? Cn : Ci;
    }
    float Aex = __shfl(Ai, (lane > 0) ? lane - 1 : 0);
    float Cex = __shfl(Ci, (lane > 0) ? lane - 1 : 0);
    if (lane == 0) { Aex = 1.0f; Cex = 0.0f; }
    float xc = Aex * x_init + Cex;  // state entering this lane's first chunk
    for (int i = 0; i < per; ++i) {
        tIn[base + i] = xc;
        xc = Ag[base + i] * xc + C[base + i];
    }
}

// ---------------- Phase B: apply carries, compute + store outputs ----------------
__global__ void kf_phaseB(const float* __restrict__ x, float* __restrict__ out,
                          float* __restrict__ ws) {
    const int lane  = threadIdx.x & 31;
    const int chunk = __builtin_amdgcn_readfirstlane(blockIdx.x * (blockDim.x >> 5) + (threadIdx.x >> 5));
    const float a = ws[WS_A], b = ws[WS_B];
    KfConsts c;
    kf_setup(a, b, lane, c);
    const int tile0  = chunk * KF_TPC;
    const int ntiles = __builtin_amdgcn_readfirstlane(min(KF_TPC, KF_TILES - tile0));
    float carry = ws[WS_TIN + chunk];
    kf_chunk<true>(x, out, tile0, ntiles, carry, c, lane);
}

extern "C" void kernel_launch(void* const* d_in, const int* in_sizes, int n_in,
                              void* d_out, int out_size, void* d_ws, size_t ws_size,
                              hipStream_t stream) {
    (void)in_sizes; (void)n_in; (void)out_size; (void)ws_size;
    const float* x  = (const float*)d_in[0];
    const float* x0 = (const float*)d_in[1];
    const float* p0 = (const float*)d_in[2];
    const float* A  = (const float*)d_in[3];
    const float* H  = (const float*)d_in[4];
    const float* Q  = (const float*)d_in[5];
    const float* R  = (const float*)d_in[6];
    float* out = (float*)d_out;
    float* ws  = (float*)d_ws;

    kf_seq_head<<<1, 1, 0, stream>>>(x, x0, p0, A, H, Q, R, out, ws);
    // 2048 chunks, 8 waves (256 threads) per block -> 256 blocks
    kf_phaseA<<<KF_CHUNKS / 8, 256, 0, stream>>>(x, ws);
    kf_scan<<<1, 32, 0, stream>>>(ws);
    kf_phaseB<<<KF_CHUNKS / 8, 256, 0, stream>>>(x, out, ws);
}